// LSTM_loss_2241972928638
// MI455X (gfx1250) — compile-verified
//
#include <hip/hip_runtime.h>

typedef float f4  __attribute__((ext_vector_type(4)));
typedef float v2f __attribute__((ext_vector_type(2)));
typedef float v8f __attribute__((ext_vector_type(8)));

#define HW 65536          // 256*256
#define CHUNK_FLOATS 8192 // per block
#define NTHREADS 256

// ---------------------------------------------------------------------------
// Kernel 1: S_all = sum over all images/pixels of output[b, 4, y, x]^2.
// Each block handles 8192 contiguous floats of one image's channel-4 slab.
// Horizontal accumulation via v_wmma_f32_16x16x4_f32 with B = ones:
//   D[m][n] += sum_k A[m][k]  (for all 16 columns n)
// => sum of all D entries = 16 * sum of A entries, independent of slot layout.
// ---------------------------------------------------------------------------
__global__ __launch_bounds__(NTHREADS) void sumsq_ch4_kernel(
    const float* __restrict__ out, float* __restrict__ partials) {
  const int blk   = blockIdx.x;
  const int slab  = blk >> 3;   // image index
  const int chunk = blk & 7;    // 8192-float chunk within 65536-float slab
  const f4* __restrict__ base =
      (const f4*)(out + ((size_t)slab * 5 + 4) * HW + (size_t)chunk * CHUNK_FLOATS);
  const int tid = threadIdx.x;

  v8f acc = {0.f, 0.f, 0.f, 0.f, 0.f, 0.f, 0.f, 0.f};
  v2f ones;
  ones[0] = 1.0f;
  ones[1] = 1.0f;

#pragma unroll
  for (int it = 0; it < CHUNK_FLOATS / (NTHREADS * 4); ++it) {
    f4 v = __builtin_nontemporal_load(&base[it * NTHREADS + tid]);
    v2f a0, a1;
    a0[0] = v.x * v.x; a0[1] = v.y * v.y;
    a1[0] = v.z * v.z; a1[1] = v.w * v.w;
    // D = A*ones + C  (chained accumulate; EXEC is all-ones: uniform control flow)
    acc = __builtin_amdgcn_wmma_f32_16x16x4_f32(false, a0, false, ones,
                                                (short)0, acc, false, false);
    acc = __builtin_amdgcn_wmma_f32_16x16x4_f32(false, a1, false, ones,
                                                (short)0, acc, false, false);
  }
  // Per-lane sum of the 8 D VGPRs; summed over lanes/block this is 16*sum(x^2).
  float t = acc[0] + acc[1] + acc[2] + acc[3] + acc[4] + acc[5] + acc[6] + acc[7];

  __shared__ float red[NTHREADS];
  red[tid] = t;
  __syncthreads();
  for (int s = NTHREADS / 2; s > 0; s >>= 1) {
    if (tid < s) red[tid] += red[tid + s];
    __syncthreads();
  }
  if (tid == 0) partials[blk] = red[0];  // carries the 16x factor
}

// ---------------------------------------------------------------------------
// Kernel 2 (one block): per-image flag via LDS atomicMax, then gather the 5
// prediction channels at each valid label cell and accumulate:
//   res[0] = sum (pred_vel - vxy)^2, res[1] = sum (pred_accel - axy)^2,
//   res[2] = sum pred_obj^2 at obj cells (correction), res[3] = n_obj (int)
// ---------------------------------------------------------------------------
__global__ __launch_bounds__(NTHREADS) void label_pass_kernel(
    const int* __restrict__ labels, int nrows,
    const float* __restrict__ out, float* __restrict__ res) {
  __shared__ int   sflag[160];
  __shared__ float rv[NTHREADS], ra[NTHREADS], ro[NTHREADS];
  __shared__ int   rc[NTHREADS];
  const int tid = threadIdx.x;

  for (int i = tid; i < 160; i += NTHREADS) sflag[i] = -0x7fffffff;
  __syncthreads();
  for (int r = tid; r < nrows; r += NTHREADS) {
    int img  = labels[r * 10 + 9];
    int flag = labels[r * 10 + 8];
    if (img >= 0 && img < 160) atomicMax(&sflag[img], flag);
  }
  __syncthreads();

  float vel = 0.f, accl = 0.f, objsq = 0.f;
  int cnt = 0;
  for (int r = tid; r < nrows; r += NTHREADS) {
    const int* row = labels + (size_t)r * 10;
    int img = row[9];
    if (img >= 1 && img < 160 && sflag[img] != -1) {
      int xc = row[0] >> 3;  // x // RES, RES = 8
      int yc = row[1] >> 3;
      const float* p = out + (size_t)(img - 1) * 5 * HW + (size_t)yc * 256 + xc;
      float p0 = p[0];
      float p1 = p[HW];
      float p2 = p[2 * HW];
      float p3 = p[3 * HW];
      float p4 = p[4 * HW];
      float vx = (float)row[4], vy = (float)row[5];
      float ax = (float)row[6], ay = (float)row[7];
      vel   += (p0 - vx) * (p0 - vx) + (p1 - vy) * (p1 - vy);
      accl  += (p2 - ax) * (p2 - ax) + (p3 - ay) * (p3 - ay);
      objsq += p4 * p4;
      cnt++;
    }
  }
  rv[tid] = vel; ra[tid] = accl; ro[tid] = objsq; rc[tid] = cnt;
  __syncthreads();
  for (int s = NTHREADS / 2; s > 0; s >>= 1) {
    if (tid < s) {
      rv[tid] += rv[tid + s];
      ra[tid] += ra[tid + s];
      ro[tid] += ro[tid + s];
      rc[tid] += rc[tid + s];
    }
    __syncthreads();
  }
  if (tid == 0) {
    res[0] = rv[0];
    res[1] = ra[0];
    res[2] = ro[0];
    ((int*)res)[3] = rc[0];
  }
}

// ---------------------------------------------------------------------------
// Kernel 3 (one block): deterministic combine -> final scalar loss.
// ---------------------------------------------------------------------------
__global__ __launch_bounds__(NTHREADS) void finalize_kernel(
    const float* __restrict__ partials, int npart,
    const float* __restrict__ res, float total_cells,
    float* __restrict__ d_out) {
  __shared__ float red[NTHREADS];
  const int tid = threadIdx.x;
  float s = 0.f;
  for (int i = tid; i < npart; i += NTHREADS) s += partials[i];
  red[tid] = s;
  __syncthreads();
  for (int st = NTHREADS / 2; st > 0; st >>= 1) {
    if (tid < st) red[tid] += red[tid + st];
    __syncthreads();
  }
  if (tid == 0) {
    float S_all = red[0] * 0.0625f;  // undo the 16x WMMA replication exactly
    float vel   = res[0];
    float accl  = res[1];
    float objsq = res[2];
    int   nobj  = ((const int*)res)[3];
    float fnobj = (float)nobj;
    float noobj_loss = (S_all - objsq) / fmaxf(total_cells - fnobj, 1.0f);
    float den  = fmaxf(2.0f * fnobj, 1.0f);
    float full = 0.5f * noobj_loss + vel / den + accl / den;
    d_out[0] = (nobj == 0) ? (0.5f * noobj_loss) : full;
  }
}

extern "C" void kernel_launch(void* const* d_in, const int* in_sizes, int n_in,
                              void* d_out, int out_size, void* d_ws, size_t ws_size,
                              hipStream_t stream) {
  const float* output = (const float*)d_in[0];
  const int*   labels = (const int*)d_in[1];
  // num_images (d_in[2]) derivable from sizes; not needed on device.

  const int nrows = in_sizes[1] / 10;           // 8256 label rows
  const int Bm1   = in_sizes[0] / (5 * HW);     // 128 images in output

  float* ws_f     = (float*)d_ws;
  float* partials = ws_f;                       // Bm1*8 floats
  float* res      = ws_f + 8192;                // 4 values

  const int nblocks = Bm1 * 8;                  // 8192 floats per block

  hipLaunchKernelGGL(sumsq_ch4_kernel, dim3(nblocks), dim3(NTHREADS), 0, stream,
                     output, partials);
  hipLaunchKernelGGL(label_pass_kernel, dim3(1), dim3(NTHREADS), 0, stream,
                     labels, nrows, output, res);
  hipLaunchKernelGGL(finalize_kernel, dim3(1), dim3(NTHREADS), 0, stream,
                     partials, nblocks, res, (float)Bm1 * (float)HW,
                     (float*)d_out);
}